// STTM_Complex_37555194036287
// MI455X (gfx1250) — compile-verified
//
#include <hip/hip_runtime.h>

typedef float v2f __attribute__((ext_vector_type(2)));
typedef float v8f __attribute__((ext_vector_type(8)));

#define NHEADS 4
#define HDIM 32

__device__ __forceinline__ v8f wmma4(v2f a, v2f b, v8f c) {
  // D = A(16x4) * B(4x16) + C, full fp32 (V_WMMA_F32_16X16X4_F32)
  return __builtin_amdgcn_wmma_f32_16x16x4_f32(false, a, false, b, (short)0, c, false, false);
}

// ---------------------------------------------------------------------------
// feat[w][n][c] = to_seq(feat_left/right): fl[w][h][c] = x[b][c][h0][w], h=2*h0+b
// ---------------------------------------------------------------------------
__global__ void __launch_bounds__(256) seq_init_kernel(const float* __restrict__ fl,
                                                       const float* __restrict__ fr,
                                                       float* __restrict__ feat) {
  long idx = (long)blockIdx.x * 256 + threadIdx.x;
  if (idx >= (long)192 * 256 * 128) return;
  int c = (int)(idx & 127);
  int n = (int)((idx >> 7) & 255);
  int w = (int)(idx >> 15);
  const float* src = fl;
  int h = n;
  if (n >= 128) { src = fr; h = n - 128; }
  int b = h & 1, h0 = h >> 1;
  feat[idx] = src[(((long)b * 128 + c) * 64 + h0) * 192 + w];
}

// ---------------------------------------------------------------------------
// Row-wise LayerNorm over 128 channels. Row m -> (w=m/xnb, n=m%xnb).
// One wave per row (wave32), float4 per lane.
// ---------------------------------------------------------------------------
__global__ void __launch_bounds__(256) ln_kernel(const float* __restrict__ X, long xws, int xnb,
                                                 float* __restrict__ Y, long yws, int ynb,
                                                 const float* __restrict__ g,
                                                 const float* __restrict__ b, int M) {
  int wid = threadIdx.x >> 5, lane = threadIdx.x & 31;
  int m = blockIdx.x * 8 + wid;
  if (m >= M) return;
  const float* xr = X + (long)(m / xnb) * xws + (long)(m % xnb) * 128;
  float4 x = *(const float4*)(xr + lane * 4);
  float s = x.x + x.y + x.z + x.w;
  for (int o = 16; o >= 1; o >>= 1) s += __shfl_xor(s, o, 32);
  float mean = s * 0.0078125f;
  float dx = x.x - mean, dy = x.y - mean, dz = x.z - mean, dw = x.w - mean;
  float q = dx * dx + dy * dy + dz * dz + dw * dw;
  for (int o = 16; o >= 1; o >>= 1) q += __shfl_xor(q, o, 32);
  float rstd = 1.0f / sqrtf(q * 0.0078125f + 1e-5f);
  float4 gg = *(const float4*)(g + lane * 4);
  float4 bb = *(const float4*)(b + lane * 4);
  float4 out;
  out.x = dx * rstd * gg.x + bb.x;
  out.y = dy * rstd * gg.y + bb.y;
  out.z = dz * rstd * gg.z + bb.z;
  out.w = dw * rstd * gg.w + bb.w;
  float* yr = Y + (long)(m / ynb) * yws + (long)(m % ynb) * 128;
  *(float4*)(yr + lane * 4) = out;
}

// ---------------------------------------------------------------------------
// Y[m, j] = sum_k X[m,k] * Wt[j,k] + bias[j] (+ R[m,j]).  K fixed = 128.
// Row m -> (w=m/nb, n=m%nb) with per-buffer strides. 64x64 macro-tile per
// workgroup, LDS-staged (stride 132 -> conflict-free A/B frags), WMMA f32.
// ---------------------------------------------------------------------------
__global__ void __launch_bounds__(256) gemm_k128_kernel(
    const float* __restrict__ X, long xws, int xnb,
    const float* __restrict__ Wt, const float* __restrict__ bias,
    float* __restrict__ Y, long yws, long yns, int ynb,
    const float* __restrict__ R, long rws, int rnb,
    int M, int N) {
  extern __shared__ float sm[];
  float* Xs = sm;             // 64 x 132
  float* Ws = sm + 64 * 132;  // 64 x 132
  const int m0 = blockIdx.x * 64;
  const int col0 = blockIdx.y * 64;
  const int tid = threadIdx.x;
  for (int idx = tid; idx < 64 * 32; idx += 256) {
    int rr = idx >> 5, c4 = (idx & 31) << 2;
    int m = m0 + rr;
    float4 v = make_float4(0.f, 0.f, 0.f, 0.f);
    if (m < M) {
      const float* xr = X + (long)(m / xnb) * xws + (long)(m % xnb) * 128;
      v = *(const float4*)(xr + c4);
    }
    *(float4*)(Xs + rr * 132 + c4) = v;
    *(float4*)(Ws + rr * 132 + c4) = *(const float4*)(Wt + (long)(col0 + rr) * 128 + c4);
  }
  __syncthreads();
  const int wid = tid >> 5, lane = tid & 31;
  const int r = lane & 15, h = lane >> 4;
  for (int t = wid; t < 16; t += 8) {   // 2 tiles per wave, uniform trip count
    const int rl = (t >> 2) * 16, cl = (t & 3) * 16;
    v8f acc = {0.f, 0.f, 0.f, 0.f, 0.f, 0.f, 0.f, 0.f};
    for (int kk = 0; kk < 128; kk += 4) {
      v2f a, bfr;
      const float* ap = Xs + (rl + r) * 132 + kk + 2 * h;
      a[0] = ap[0]; a[1] = ap[1];
      const float* bp = Ws + (cl + r) * 132 + kk + 2 * h;
      bfr[0] = bp[0]; bfr[1] = bp[1];
      acc = wmma4(a, bfr, acc);
    }
    for (int j = 0; j < 8; j++) {
      int m = m0 + rl + j + 8 * h;
      if (m < M) {
        int cg = col0 + cl + r;
        float val = acc[j] + bias[cg];
        if (R) val += R[(long)(m / rnb) * rws + (long)(m % rnb) * 128 + cg];
        Y[(long)(m / ynb) * yws + (long)(m % ynb) * yns + cg] = val;
      }
    }
  }
}

// ---------------------------------------------------------------------------
// Fused relative-position attention for one (head e, batch n).
//   S[w,v] = Q[w]·K[v] + Q[w]·projK[p] + K[v]·projQ[p],  p = rev?191+w-v:191-w+v
//   P = softmax_v(S * scale);  O = P @ V.
// All three score terms computed with fp32 WMMA; the two Toeplitz terms via
// 16x384 band GEMMs against the positional projection, then diagonal
// scatter-add. Whole 192x193 score matrix lives in LDS (320,256 B <= 320 KB).
// ---------------------------------------------------------------------------
__global__ void __launch_bounds__(256) attn_kernel(
    const float* __restrict__ qb, const float* __restrict__ kb,
    const float* __restrict__ vb, const float* __restrict__ proj,
    float* __restrict__ ob, int nb, int rev, float scale) {
  extern __shared__ float sm[];
  float* QsT  = sm;                  // 32 x 193  (Q^T, 193 = conflict-free)
  float* KsT  = QsT + 32 * 193;      // 32 x 193  (K^T)
  float* S    = KsT + 32 * 193;      // 192 x 193 (padded for P A-frags)
  float* pqT  = S + 192 * 193;       // 32 x 383  (projQ head slice, transposed)
  float* pkT  = pqT + 32 * 383;      // 32 x 383  (projK head slice, transposed)
  float* band = pkT + 32 * 383;      // 16 x 384  (Toeplitz band staging)
  const int e = blockIdx.x, n = blockIdx.y;
  const int tid = threadIdx.x;
  const int co = e * HDIM;
  for (int idx = tid; idx < 192 * 32; idx += 256) {
    int w = idx >> 5, c = idx & 31;
    long off = ((long)w * nb + n) * 128 + co + c;
    QsT[c * 193 + w] = qb[off];
    KsT[c * 193 + w] = kb[off];
  }
  for (int idx = tid; idx < 383 * 32; idx += 256) {
    int p = idx >> 5, c = idx & 31;
    pqT[c * 383 + p] = proj[(long)p * 256 + co + c];
    pkT[c * 383 + p] = proj[(long)p * 256 + 128 + co + c];
  }
  __syncthreads();
  const int wid = tid >> 5, lane = tid & 31;
  const int r = lane & 15, h = lane >> 4;
  // ---- Term A: S = Q @ K^T (144 16x16 tiles, 18 per wave) ----
  for (int t = wid; t < 144; t += 8) {
    const int w0 = (t / 12) * 16, v0 = (t % 12) * 16;
    v8f acc = {0.f, 0.f, 0.f, 0.f, 0.f, 0.f, 0.f, 0.f};
    for (int kk = 0; kk < 32; kk += 4) {
      v2f a, bfr;
      a[0] = QsT[(kk + 2 * h) * 193 + w0 + r];
      a[1] = QsT[(kk + 2 * h + 1) * 193 + w0 + r];
      bfr[0] = KsT[(kk + 2 * h) * 193 + v0 + r];
      bfr[1] = KsT[(kk + 2 * h + 1) * 193 + v0 + r];
      acc = wmma4(a, bfr, acc);
    }
    for (int j = 0; j < 8; j++) S[(w0 + j + 8 * h) * 193 + v0 + r] = acc[j];
  }
  __syncthreads();
  // ---- Term B: band = Q[w0:w0+16] @ projK^T, then S[w,v] += band[wi, p(w,v)] ----
  for (int wb = 0; wb < 12; wb++) {
    const int w0 = wb * 16;
    for (int t = wid; t < 24; t += 8) {       // 24 column tiles of 16
      const int p0 = t * 16;
      v8f acc = {0.f, 0.f, 0.f, 0.f, 0.f, 0.f, 0.f, 0.f};
      for (int kk = 0; kk < 32; kk += 4) {
        v2f a, bfr;
        a[0] = QsT[(kk + 2 * h) * 193 + w0 + r];
        a[1] = QsT[(kk + 2 * h + 1) * 193 + w0 + r];
        bfr[0] = pkT[(kk + 2 * h) * 383 + p0 + r];
        bfr[1] = pkT[(kk + 2 * h + 1) * 383 + p0 + r];
        acc = wmma4(a, bfr, acc);
      }
      for (int j = 0; j < 8; j++) band[(j + 8 * h) * 384 + p0 + r] = acc[j];
    }
    __syncthreads();
    for (int idx = tid; idx < 16 * 192; idx += 256) {
      int wi = idx / 192, v = idx - wi * 192;
      int w = w0 + wi;
      int p = rev ? (191 + w - v) : (191 - w + v);
      S[w * 193 + v] += band[wi * 384 + p];
    }
    __syncthreads();
  }
  // ---- Term C: band = K[v0:v0+16] @ projQ^T, then S[w,v] += band[vi, p(w,v)] ----
  for (int vbk = 0; vbk < 12; vbk++) {
    const int v0 = vbk * 16;
    for (int t = wid; t < 24; t += 8) {
      const int p0 = t * 16;
      v8f acc = {0.f, 0.f, 0.f, 0.f, 0.f, 0.f, 0.f, 0.f};
      for (int kk = 0; kk < 32; kk += 4) {
        v2f a, bfr;
        a[0] = KsT[(kk + 2 * h) * 193 + v0 + r];
        a[1] = KsT[(kk + 2 * h + 1) * 193 + v0 + r];
        bfr[0] = pqT[(kk + 2 * h) * 383 + p0 + r];
        bfr[1] = pqT[(kk + 2 * h + 1) * 383 + p0 + r];
        acc = wmma4(a, bfr, acc);
      }
      for (int j = 0; j < 8; j++) band[(j + 8 * h) * 384 + p0 + r] = acc[j];
    }
    __syncthreads();
    for (int idx = tid; idx < 16 * 192; idx += 256) {
      int vi = idx / 192, w = idx - vi * 192;
      int v = v0 + vi;
      int p = rev ? (191 + w - v) : (191 - w + v);
      S[w * 193 + v] += band[vi * 384 + p];
    }
    __syncthreads();
  }
  // ---- Row softmax (scale folded in; one wave per row, 6 elems/lane) ----
  for (int row = wid; row < 192; row += 8) {
    float vals[6];
    float mx = -3.4e38f;
    for (int i = 0; i < 6; i++) {
      vals[i] = S[row * 193 + lane + i * 32] * scale;
      mx = fmaxf(mx, vals[i]);
    }
    for (int o = 16; o >= 1; o >>= 1) mx = fmaxf(mx, __shfl_xor(mx, o, 32));
    float sum = 0.f;
    for (int i = 0; i < 6; i++) { vals[i] = expf(vals[i] - mx); sum += vals[i]; }
    for (int o = 16; o >= 1; o >>= 1) sum += __shfl_xor(sum, o, 32);
    float inv = 1.f / sum;
    for (int i = 0; i < 6; i++) S[row * 193 + lane + i * 32] = vals[i] * inv;
  }
  __syncthreads();
  // ---- O = P @ V (24 16x16 tiles; V B-frags straight from L2/L1) ----
  for (int t = wid; t < 24; t += 8) {
    const int w0 = (t >> 1) * 16, c0 = (t & 1) * 16;
    v8f acc = {0.f, 0.f, 0.f, 0.f, 0.f, 0.f, 0.f, 0.f};
    for (int kk = 0; kk < 192; kk += 4) {
      v2f a, bfr;
      const float* ap = S + (w0 + r) * 193 + kk + 2 * h;
      a[0] = ap[0]; a[1] = ap[1];
      long vr = ((long)(kk + 2 * h) * nb + n) * 128 + co + c0 + r;
      bfr[0] = vb[vr];
      bfr[1] = vb[vr + (long)nb * 128];
      acc = wmma4(a, bfr, acc);
    }
    for (int j = 0; j < 8; j++)
      ob[((long)(w0 + j + 8 * h) * nb + n) * 128 + co + c0 + r] = acc[j];
  }
}

// ---------------------------------------------------------------------------
extern "C" void kernel_launch(void* const* d_in, const int* in_sizes, int n_in,
                              void* d_out, int out_size, void* d_ws, size_t ws_size,
                              hipStream_t stream) {
  (void)in_sizes; (void)n_in; (void)out_size; (void)ws_size;
  const float* feat_left  = (const float*)d_in[0];
  const float* feat_right = (const float*)d_in[1];
  const float* pos_enc    = (const float*)d_in[2];
  const float* s_iw = (const float*)d_in[3];
  const float* s_ib = (const float*)d_in[4];
  const float* s_ow = (const float*)d_in[5];
  const float* s_ob = (const float*)d_in[6];
  const float* s_g  = (const float*)d_in[7];
  const float* s_b  = (const float*)d_in[8];
  const float* c_iw = (const float*)d_in[9];
  const float* c_ib = (const float*)d_in[10];
  const float* c_ow = (const float*)d_in[11];
  const float* c_ob = (const float*)d_in[12];
  const float* c_g1 = (const float*)d_in[13];
  const float* c_b1 = (const float*)d_in[14];
  const float* c_g2 = (const float*)d_in[15];
  const float* c_b2 = (const float*)d_in[16];

  float* feat = (float*)d_out;            // live activations, (192,256,128)
  float* ws = (float*)d_ws;
  const long FEATN = (long)192 * 256 * 128;
  float* f2   = ws;                       // LN output
  float* qb   = f2 + FEATN;
  float* kb   = qb + FEATN;
  float* vb   = kb + FEATN;
  float* obuf = vb + FEATN;
  float* proj = obuf + FEATN;             // 383 x 256 positional projection

  const size_t attn_smem = (size_t)(2 * 32 * 193 + 192 * 193 + 2 * 32 * 383 + 16 * 384) * 4; // 320256 B
  const size_t gemm_smem = (size_t)(2 * 64 * 132) * 4;                                       // 67584 B
  (void)hipFuncSetAttribute(reinterpret_cast<const void*>(attn_kernel),
                            hipFuncAttributeMaxDynamicSharedMemorySize, (int)attn_smem);
  (void)hipFuncSetAttribute(reinterpret_cast<const void*>(gemm_k128_kernel),
                            hipFuncAttributeMaxDynamicSharedMemorySize, (int)gemm_smem);

  const float scale = 0.17677669529663687f;  // 1/sqrt(32)
  const long WS256 = 256L * 128;  // w-stride of a (w,256,128) buffer
  const long WS128 = 128L * 128;  // w-stride of a (w,128,128) buffer
  const long HALF  = 128L * 128;  // offset of right half inside a w-block
  const int  SELF_M = 192 * 256, CROSS_M = 192 * 128;

  auto GEMM = [&](const float* X, long xws, int xnb,
                  const float* Wt, const float* bias,
                  float* Y, long yws, long yns, int ynb,
                  const float* R, long rws, int rnb, int M, int N) {
    dim3 g((unsigned)((M + 63) / 64), (unsigned)(N / 64));
    gemm_k128_kernel<<<g, 256, gemm_smem, stream>>>(X, xws, xnb, Wt, bias,
                                                    Y, yws, yns, ynb, R, rws, rnb, M, N);
  };
  auto LN = [&](const float* X, long xws, int xnb, float* Y, long yws, int ynb,
                const float* gg, const float* bb, int M) {
    ln_kernel<<<dim3((unsigned)((M + 7) / 8)), 256, 0, stream>>>(X, xws, xnb, Y, yws, ynb, gg, bb, M);
  };
  auto ATTN = [&](const float* q_, const float* k_, const float* v_, float* o_, int nb, int rev) {
    attn_kernel<<<dim3(NHEADS, (unsigned)nb), 256, attn_smem, stream>>>(q_, k_, v_, proj, o_, nb, rev, scale);
  };

  // feat = concat(to_seq(left), to_seq(right))
  seq_init_kernel<<<dim3((unsigned)((FEATN + 255) / 256)), 256, 0, stream>>>(feat_left, feat_right, feat);

  for (int i = 0; i < 4; i++) {
    const float* siw = s_iw + (long)i * 384 * 128;
    const float* sib = s_ib + (long)i * 384;
    const float* sow = s_ow + (long)i * 128 * 128;
    const float* sob = s_ob + (long)i * 128;
    const float* ciw = c_iw + (long)i * 384 * 128;
    const float* cib = c_ib + (long)i * 384;
    const float* cow = c_ow + (long)i * 128 * 128;
    const float* cob = c_ob + (long)i * 128;

    // ---------------- self-attention ----------------
    LN(feat, WS256, 256, f2, WS256, 256, s_g + i * 128, s_b + i * 128, SELF_M);
    GEMM(pos_enc, 0, 1023, siw, sib, proj, 0, 256, 1023, nullptr, 0, 1, 383, 256);   // pos projection
    GEMM(f2, WS256, 256, siw,             sib,       qb, WS256, 128, 256, nullptr, 0, 1, SELF_M, 128);
    GEMM(f2, WS256, 256, siw + 128 * 128, sib + 128, kb, WS256, 128, 256, nullptr, 0, 1, SELF_M, 128);
    GEMM(f2, WS256, 256, siw + 256 * 128, sib + 256, vb, WS256, 128, 256, nullptr, 0, 1, SELF_M, 128);
    ATTN(qb, kb, vb, obuf, 256, 0);
    GEMM(obuf, WS256, 256, sow, sob, feat, WS256, 128, 256, feat, WS256, 256, SELF_M, 128); // +residual

    // ---------------- cross A: fr attends to fl (pos_flip -> rev=1) ----------------
    LN(feat, WS256, 256, f2, WS256, 256, c_g1 + i * 128, c_b1 + i * 128, SELF_M);    // [fl2 | fr2]
    GEMM(pos_enc, 0, 1023, ciw, cib, proj, 0, 256, 1023, nullptr, 0, 1, 383, 256);
    GEMM(f2 + HALF, WS256, 128, ciw,             cib,       qb, WS128, 128, 128, nullptr, 0, 1, CROSS_M, 128); // q=fr2
    GEMM(f2,        WS256, 128, ciw + 128 * 128, cib + 128, kb, WS128, 128, 128, nullptr, 0, 1, CROSS_M, 128); // k=fl2
    GEMM(f2,        WS256, 128, ciw + 256 * 128, cib + 256, vb, WS128, 128, 128, nullptr, 0, 1, CROSS_M, 128); // v=fl2
    ATTN(qb, kb, vb, obuf, 128, 1);
    GEMM(obuf, WS128, 128, cow, cob, feat + HALF, WS256, 128, 128, feat + HALF, WS256, 128, CROSS_M, 128); // fr += ...

    // ---------------- cross B: fl attends to fr_new (pos_enc -> rev=0) ----------------
    LN(feat + HALF, WS256, 128, f2 + HALF, WS256, 128, c_g2 + i * 128, c_b2 + i * 128, CROSS_M); // fr2n
    GEMM(f2,        WS256, 128, ciw,             cib,       qb, WS128, 128, 128, nullptr, 0, 1, CROSS_M, 128); // q=fl2
    GEMM(f2 + HALF, WS256, 128, ciw + 128 * 128, cib + 128, kb, WS128, 128, 128, nullptr, 0, 1, CROSS_M, 128); // k=fr2n
    GEMM(f2 + HALF, WS256, 128, ciw + 256 * 128, cib + 256, vb, WS128, 128, 128, nullptr, 0, 1, CROSS_M, 128); // v=fr2n
    ATTN(qb, kb, vb, obuf, 128, 0);
    GEMM(obuf, WS128, 128, cow, cob, feat, WS256, 128, 128, feat, WS256, 128, CROSS_M, 128); // fl += ...
  }
}